// FBGCN_36180804502072
// MI455X (gfx1250) — compile-verified
//
#include <hip/hip_runtime.h>
#include <hip/hip_bf16.h>

typedef float v2f __attribute__((ext_vector_type(2)));
typedef float v8f __attribute__((ext_vector_type(8)));

// ---------------------------------------------------------------------------
// GCN normalization kernels
// ---------------------------------------------------------------------------

__global__ void k_fill1(float* __restrict__ p, long long n) {
  long long i = (long long)blockIdx.x * blockDim.x + threadIdx.x;
  for (; i < n; i += (long long)gridDim.x * blockDim.x) p[i] = 1.0f;
}

__global__ void k_degacc(const long long* __restrict__ ei,
                         const float* __restrict__ ew,
                         float* __restrict__ deg, long long E) {
  long long i = (long long)blockIdx.x * blockDim.x + threadIdx.x;
  for (; i < E; i += (long long)gridDim.x * blockDim.x) {
    long long d = ei[E + i];
    atomicAdd(&deg[d], ew[i]);
  }
}

__global__ void k_rsqrt_inplace(float* __restrict__ p, long long n) {
  long long i = (long long)blockIdx.x * blockDim.x + threadIdx.x;
  for (; i < n; i += (long long)gridDim.x * blockDim.x) {
    float d = p[i];
    p[i] = d > 0.0f ? rsqrtf(d) : 0.0f;
  }
}

__global__ void k_norm(const long long* __restrict__ ei,
                       const float* __restrict__ ew,
                       const float* __restrict__ dinv,
                       float* __restrict__ norm, long long E) {
  long long i = (long long)blockIdx.x * blockDim.x + threadIdx.x;
  for (; i < E; i += (long long)gridDim.x * blockDim.x) {
    long long s = ei[i];
    long long d = ei[E + i];
    norm[i] = dinv[s] * ew[i] * dinv[d];
  }
}

// ---------------------------------------------------------------------------
// Weight pre-pack: rearrange W[K, ncols] into the per-lane wave32 B-fragment
// order for V_WMMA_F32_16X16X4_F32, including zero-fill for padded columns.
//   Wp[((ks*NT + t)*32 + lane)*2 + j] = W[(4*ks + 2*(lane>>4) + j)*ncols + t*16 + (lane&15)]
// In the GEMM a lane then loads one b64 per (ks, t) at base + immediate.
// ---------------------------------------------------------------------------

__global__ void k_packW(const float* __restrict__ W, float* __restrict__ Wp,
                        int K, int ncols, int NT) {
  const int total = (K / 4) * NT * 64;               // *32 lanes * 2 floats
  for (int i = blockIdx.x * blockDim.x + threadIdx.x; i < total;
       i += gridDim.x * blockDim.x) {
    const int j    = i & 1;
    const int lane = (i >> 1) & 31;
    const int t    = (i >> 6) % NT;
    const int ks   = (i >> 6) / NT;
    const int half = lane >> 4;
    const int l    = lane & 15;
    const int ka   = 4 * ks + 2 * half + j;
    const int col  = t * 16 + l;
    Wp[i] = (col < ncols) ? W[ka * ncols + col] : 0.0f;
  }
}

// ---------------------------------------------------------------------------
// fp32 WMMA GEMM: C[N, NCOLS] = relu?(A[N, K]) @ W[K, NCOLS]
// One wave32 per (16-row x NT*16-col) strip. K compile-time -> fully unrolled:
// per k-step 1 A load (b64) + NT packed-B loads (b64, base+imm) + NT v_wmma.
// ISA 7.12.2 layouts (wave32):
//   A 16x4 : lane l<16 -> M=l ; VGPRj = K = k + 2*half + j
//   B 4x16 : VGPRj, col = lane&15, row K = k + 2*half + j   (pre-packed)
//   C 16x16: VGPRv -> M = row0 + 8*half + v, N = col0 + (lane&15)
// ---------------------------------------------------------------------------

template <int NT, int K, int NCOLS, bool RELU>
__global__ __launch_bounds__(32)
void k_gemm_wmma(const float* __restrict__ A, const float* __restrict__ Wp,
                 float* __restrict__ C, int Nrows) {
  const int row0 = blockIdx.x * 16;
  const int lane = threadIdx.x;       // 0..31
  const int half = lane >> 4;         // 0 | 1
  const int l    = lane & 15;

  const int row  = row0 + l;
  const int rowc = row < Nrows ? row : (Nrows - 1);   // clamp; masked at store
  const float* __restrict__ Arow = A + (size_t)rowc * K + 2 * half;
  const float* __restrict__ Bp   = Wp + lane * 2;

  v8f acc[NT] = {};
#pragma unroll
  for (int ks = 0; ks < K / 4; ++ks) {
    v2f a = *(const v2f*)(Arow + 4 * ks);
    if (RELU) { a.x = fmaxf(a.x, 0.0f); a.y = fmaxf(a.y, 0.0f); }
#pragma unroll
    for (int t = 0; t < NT; ++t) {
      const v2f b = *(const v2f*)(Bp + (ks * NT + t) * 64);
      acc[t] = __builtin_amdgcn_wmma_f32_16x16x4_f32(
          /*neg_a=*/false, a, /*neg_b=*/false, b,
          /*c_mod=*/(short)0, acc[t], /*reuse_a=*/false, /*reuse_b=*/false);
    }
  }

  const int rbase = row0 + half * 8;
#pragma unroll
  for (int t = 0; t < NT; ++t) {
    const int col = t * 16 + l;
    if (col < NCOLS) {
#pragma unroll
      for (int v = 0; v < 8; ++v) {
        const int r = rbase + v;
        if (r < Nrows) C[(size_t)r * NCOLS + col] = acc[t][v];
      }
    }
  }
}

// ---------------------------------------------------------------------------
// Aggregation. agg[n][f] = dinv[n]^2*hw[n][f] + b[f] (self-loop + bias, full
// overwrite -> also clears poison), then float4-chunked atomic edge scatter.
// F in {64,32,4}; one thread handles a 4-feature chunk.
// ---------------------------------------------------------------------------

__global__ void k_selfinit4(const float* __restrict__ dinv,
                            const float* __restrict__ hw,
                            const float* __restrict__ bias,
                            float* __restrict__ agg,
                            long long N, int fshift) {
  const int cshift = fshift - 2;                 // chunks per node = F/4
  const long long total = N << cshift;
  long long i = (long long)blockIdx.x * blockDim.x + threadIdx.x;
  for (; i < total; i += (long long)gridDim.x * blockDim.x) {
    const long long n = i >> cshift;
    const int c = (int)(i & ((1 << cshift) - 1));
    const float di = dinv[n];
    const float d2 = di * di;
    const float4 v  = *(const float4*)(hw + (n << fshift) + 4 * c);
    const float4 bb = *(const float4*)(bias + 4 * c);
    float4 r;
    r.x = d2 * v.x + bb.x;
    r.y = d2 * v.y + bb.y;
    r.z = d2 * v.z + bb.z;
    r.w = d2 * v.w + bb.w;
    *(float4*)(agg + (n << fshift) + 4 * c) = r;
  }
}

__global__ void k_scatter4(const long long* __restrict__ ei,
                           const float* __restrict__ norm,
                           const float* __restrict__ hw,
                           float* __restrict__ agg,
                           long long E, int fshift) {
  const int cshift = fshift - 2;                 // chunks per edge = F/4
  const long long total = E << cshift;
  long long i = (long long)blockIdx.x * blockDim.x + threadIdx.x;
  for (; i < total; i += (long long)gridDim.x * blockDim.x) {
    const long long e = i >> cshift;
    const int c = (int)(i & ((1 << cshift) - 1));
    const long long s = ei[e];
    const long long d = ei[E + e];
    const float nrm = norm[e];
    const float4 v = *(const float4*)(hw + (s << fshift) + 4 * c);
    float* dst = agg + (d << fshift) + 4 * c;
    atomicAdd(dst + 0, nrm * v.x);
    atomicAdd(dst + 1, nrm * v.y);
    atomicAdd(dst + 2, nrm * v.z);
    atomicAdd(dst + 3, nrm * v.w);
  }
}

// ---------------------------------------------------------------------------
// Launch
// ---------------------------------------------------------------------------

static inline unsigned blocks_for(long long total, int tpb) {
  long long b = (total + tpb - 1) / tpb;
  if (b > (1ll << 20)) b = (1ll << 20);
  if (b < 1) b = 1;
  return (unsigned)b;
}

extern "C" void kernel_launch(void* const* d_in, const int* in_sizes, int n_in,
                              void* d_out, int out_size, void* d_ws, size_t ws_size,
                              hipStream_t stream) {
  const float*     x  = (const float*)d_in[0];      // [N,128]
  const long long* ei = (const long long*)d_in[1];  // [2,E] int64
  const float*     ew = (const float*)d_in[2];      // [E]
  const float*     W1 = (const float*)d_in[3];      // [128,64]
  const float*     b1 = (const float*)d_in[4];      // [64]
  const float*     W2 = (const float*)d_in[5];      // [64,32]
  const float*     b2 = (const float*)d_in[6];      // [32]
  const float*     W3 = (const float*)d_in[7];      // [32,4]
  const float*     b3 = (const float*)d_in[8];      // [4]

  const long long N = in_sizes[0] / 128;
  const long long E = in_sizes[2];
  float* out = (float*)d_out;                       // [N,4]

  // ---- workspace carve-out (256B aligned) ----
  char* ws = (char*)d_ws;
  auto carve = [&](size_t bytes) {
    char* p = ws;
    ws += (bytes + 255) & ~(size_t)255;
    return p;
  };
  float* dinv = (float*)carve((size_t)N * 4);        // degree -> rsqrt in place
  float* norm = (float*)carve((size_t)E * 4);        // per-edge norm
  float* bufA = (float*)carve((size_t)N * 64 * 4);   // hw (gemm output)
  float* bufB = (float*)carve((size_t)N * 64 * 4);   // agg / h
  float* Wp1  = (float*)carve((size_t)(128 / 4) * 4 * 64 * 4);  // 8192 floats
  float* Wp2  = (float*)carve((size_t)(64 / 4) * 2 * 64 * 4);   // 2048 floats
  float* Wp3  = (float*)carve((size_t)(32 / 4) * 1 * 64 * 4);   // 512 floats

  const int TPB = 256;

  // ---- weight pre-pack (tiny, once per call) ----
  k_packW<<<32, TPB, 0, stream>>>(W1, Wp1, 128, 64, 4);
  k_packW<<<8, TPB, 0, stream>>>(W2, Wp2, 64, 32, 2);
  k_packW<<<2, TPB, 0, stream>>>(W3, Wp3, 32, 4, 1);

  // ---- gcn_norm (computed once, reused by all 3 layers) ----
  k_fill1<<<blocks_for(N, TPB), TPB, 0, stream>>>(dinv, N);   // self-loop wt 1
  k_degacc<<<blocks_for(E, TPB), TPB, 0, stream>>>(ei, ew, dinv, E);
  k_rsqrt_inplace<<<blocks_for(N, TPB), TPB, 0, stream>>>(dinv, N);
  k_norm<<<blocks_for(E, TPB), TPB, 0, stream>>>(ei, ew, dinv, norm, E);

  const unsigned rowTiles = (unsigned)((N + 15) / 16);

  // ---- layer 1: 128 -> 64 (ReLU fused into layer-2 GEMM load) ----
  k_gemm_wmma<4, 128, 64, false><<<rowTiles, 32, 0, stream>>>(x, Wp1, bufA, (int)N);
  k_selfinit4<<<blocks_for(N << 4, TPB), TPB, 0, stream>>>(dinv, bufA, b1, bufB, N, 6);
  k_scatter4<<<blocks_for(E << 4, TPB), TPB, 0, stream>>>(ei, norm, bufA, bufB, E, 6);

  // ---- layer 2: 64 -> 32 (input = relu(bufB)) ----
  k_gemm_wmma<2, 64, 32, true><<<rowTiles, 32, 0, stream>>>(bufB, Wp2, bufA, (int)N);
  k_selfinit4<<<blocks_for(N << 3, TPB), TPB, 0, stream>>>(dinv, bufA, b2, bufB, N, 5);
  k_scatter4<<<blocks_for(E << 3, TPB), TPB, 0, stream>>>(ei, norm, bufA, bufB, E, 5);

  // ---- layer 3: 32 -> 4 (input = relu(bufB)), straight into d_out ----
  k_gemm_wmma<1, 32, 4, true><<<rowTiles, 32, 0, stream>>>(bufB, Wp3, bufA, (int)N);
  k_selfinit4<<<blocks_for(N, TPB), TPB, 0, stream>>>(dinv, bufA, b3, out, N, 2);
  k_scatter4<<<blocks_for(E, TPB), TPB, 0, stream>>>(ei, norm, bufA, out, E, 2);
}